// MovementPatternEncoder_78237124264597
// MI455X (gfx1250) — compile-verified
//
#include <hip/hip_runtime.h>
#include <hip/hip_bf16.h>

// ---------------------------------------------------------------------------
// MovementPatternEncoder for MI455X (gfx1250, wave32, WMMA)
//
// K1: per-row bigram histogram + entropy (memory-bound, ~64MB @ 23.3TB/s)
// K2: 21-node GAT + readout MLP in ONE workgroup using V_WMMA_F32_16X16X4_F32
// K3: rank-1 broadcast  out[b][j] = const[j] + me[b]*dir[j]
// ---------------------------------------------------------------------------

#define NNODE 21
#define NBINS (NNODE * NNODE) // 441
#define DMODEL 128

typedef __attribute__((ext_vector_type(2))) float v2f;
typedef __attribute__((ext_vector_type(8))) float v8f;

// ---------------------------------------------------------------------------
// K0: zero the global-mask scratch (must happen every call: ws is not
// re-zeroed by the harness and first call sees 0xAA poison)
// ---------------------------------------------------------------------------
__global__ void zero_ws_kernel(int* gmask) {
    int t = threadIdx.x;
    if (t < NBINS) gmask[t] = 0;
}

// ---------------------------------------------------------------------------
// K1: one block per batch row. LDS 441-bin histogram, entropy, global mask.
// int4 vector loads; S is a multiple of 4 so only the very last pair of the
// last quad needs a guard.
// ---------------------------------------------------------------------------
__global__ void hist_entropy_kernel(const int* __restrict__ ids,
                                    float* __restrict__ me_out,
                                    int* __restrict__ gmask,
                                    int S) {
    __shared__ unsigned int hist[NBINS];
    __shared__ float red[256];

    const int b = blockIdx.x;
    const int t = threadIdx.x;

    for (int i = t; i < NBINS; i += 256) hist[i] = 0u;
    __syncthreads();

    const int* __restrict__ row = ids + (long long)b * S;
    const int4* __restrict__ row4 = (const int4*)row;
    const int nq = S >> 2;

    for (int q = t; q < nq; q += 256) {
        int4 v = row4[q];
        int i0 = q << 2;
        int ni = i0 + 4;                      // clamp: unconditional load,
        int nxt = row[ni < S ? ni : S - 1];   // value unused when clamped
        atomicAdd(&hist[v.x * NNODE + v.y], 1u);
        atomicAdd(&hist[v.y * NNODE + v.z], 1u);
        atomicAdd(&hist[v.z * NNODE + v.w], 1u);
        if (q < nq - 1) atomicAdd(&hist[v.w * NNODE + nxt], 1u);
    }
    __syncthreads();

    // global edge-exists mask (idempotent OR)
    for (int i = t; i < NBINS; i += 256)
        if (hist[i]) atomicOr(&gmask[i], 1);

    // entropy: me = -sum p*log(p+1e-10), p = cnt/(S-1); cnt==0 contributes 0
    const float inv = 1.0f / (float)(S - 1);
    float part = 0.0f;
    for (int i = t; i < NBINS; i += 256) {
        float p = (float)hist[i] * inv;
        part += p * logf(p + 1e-10f);
    }
    red[t] = part;
    __syncthreads();
    for (int off = 128; off > 0; off >>= 1) {
        if (t < off) red[t] += red[t + off];
        __syncthreads();
    }
    if (t == 0) me_out[b] = -red[0];
}

// ---------------------------------------------------------------------------
// K2: single workgroup (256 threads = 8 wave32). All GEMMs via WMMA f32.
// A-tile rows >= 21 are address-clamped (NOT zeroed): those C rows are never
// stored, and row independence of D = A*B + C makes their values irrelevant.
// This keeps every fragment load unconditional (no exec-mask branching).
// ---------------------------------------------------------------------------
__global__ void __launch_bounds__(256)
graph_kernel(const float* __restrict__ embed,   // 21 x 128
             const float* __restrict__ W1,      // 128 x 256
             const float* __restrict__ a_src1,  // 4 x 64
             const float* __restrict__ a_dst1,  // 4 x 64
             const float* __restrict__ b1,      // 256
             const float* __restrict__ W2,      // 256 x 128
             const float* __restrict__ a_src2,  // 1 x 128
             const float* __restrict__ a_dst2,  // 1 x 128
             const float* __restrict__ b2,      // 128
             const float* __restrict__ We,      // 1 x 32
             const float* __restrict__ be,      // 32
             const float* __restrict__ Wr1,     // 2688 x 256
             const float* __restrict__ br1,     // 256
             const float* __restrict__ Wr2,     // 256 x 128
             const float* __restrict__ br2,     // 128
             const float* __restrict__ Wo,      // 160 x 128
             const float* __restrict__ bo,      // 128
             const int* __restrict__ gmask,     // 441
             float* __restrict__ cvec,          // out: 128
             float* __restrict__ dvec) {        // out: 128
    // bufA: xw1 (21x256), later reused: xw2 (21x128) at [0], h (21x128) at [2688]
    __shared__ float bufA[NNODE * 256];
    __shared__ float bufB[NNODE * 256]; // h1 (21x256)
    __shared__ float s1l[NNODE * 4], d1l[NNODE * 4];
    __shared__ float s2l[NNODE], d2l[NNODE];
    __shared__ float hg1[256];
    __shared__ float hgr[DMODEL];
    __shared__ int maskl[NBINS];

    const int t = threadIdx.x;
    const int wave = t >> 5;
    const int lane = t & 31;
    const int half = lane >> 4; // 0: lanes 0-15, 1: lanes 16-31
    const int l = lane & 15;

    for (int i = t; i < NBINS; i += 256) maskl[i] = gmask[i];
    __syncthreads();

    // ---- GEMM1: xw1 = embed(21x128) @ W1(128x256), WMMA f32 16x16x4 -------
    // 2 M-tiles x 16 N-tiles = 32 tiles, 4 per wave
    for (int it = 0; it < 4; ++it) {
        int tile = wave + 8 * it;
        int mt = tile >> 4, nt = tile & 15;
        v8f acc = {0.f, 0.f, 0.f, 0.f, 0.f, 0.f, 0.f, 0.f};
        int arow = mt * 16 + l;
        int ar = (arow < NNODE) ? arow : 0;   // address clamp only
        int bcol = nt * 16 + l;
        const float* Abase = embed + ar * 128 + 2 * half;
        for (int k = 0; k < 128; k += 4) {
            // A 16x4: lanes 0-15 -> K=k,k+1 ; lanes 16-31 -> K=k+2,k+3
            v2f a = *(const v2f*)(Abase + k);          // global_load_b64
            v2f bf;
            bf.x = W1[(k + 2 * half + 0) * 256 + bcol];
            bf.y = W1[(k + 2 * half + 1) * 256 + bcol];
            acc = __builtin_amdgcn_wmma_f32_16x16x4_f32(
                false, a, false, bf, (short)0, acc, false, false);
        }
        // C/D: VGPR g -> M=g (lanes 0-15) / M=g+8 (lanes 16-31)
        for (int g = 0; g < 8; ++g) {
            int row = mt * 16 + g + half * 8;
            if (row < NNODE) bufA[row * 256 + nt * 16 + l] = acc[g];
        }
    }
    __syncthreads();

    // ---- GAT1 attention logits: s/d per (node, head) -----------------------
    if (t < NNODE * 4) {
        int n = t >> 2, h = t & 3;
        float s = 0.f, d = 0.f;
        for (int c = 0; c < 64; ++c) {
            float x = bufA[n * 256 + h * 64 + c];
            s += x * a_src1[h * 64 + c];
            d += x * a_dst1[h * 64 + c];
        }
        s1l[t] = s;
        d1l[t] = d;
    }
    __syncthreads();

    // ---- GAT1 softmax over sources + aggregate + ELU -> h1 (bufB) ----------
    if (t < NNODE * 4) {
        int j = t >> 2, h = t & 3;
        float dj = d1l[j * 4 + h];
        float ev[NNODE];
        float mx = -1e30f;
        for (int i = 0; i < NNODE; ++i) {
            float e = s1l[i * 4 + h] + dj;
            e = (e >= 0.f) ? e : 0.2f * e; // leaky_relu(0.2)
            bool m = (maskl[i * NNODE + j] != 0) || (i == j);
            ev[i] = m ? e : -1e30f;
            mx = fmaxf(mx, ev[i]);
        }
        float ssum = 0.f;
        for (int i = 0; i < NNODE; ++i) {
            float x = (ev[i] > -1e29f) ? expf(ev[i] - mx) : 0.f;
            ev[i] = x;
            ssum += x;
        }
        float rs = 1.f / ssum;
        for (int c = 0; c < 64; ++c) {
            float acc = 0.f;
            for (int i = 0; i < NNODE; ++i)
                acc += ev[i] * bufA[i * 256 + h * 64 + c];
            float o = acc * rs + b1[h * 64 + c];
            bufB[j * 256 + h * 64 + c] = (o > 0.f) ? o : expm1f(o); // ELU
        }
    }
    __syncthreads();

    // ---- GEMM2: xw2 = h1(21x256) @ W2(256x128), WMMA f32 -------------------
    // 2 M-tiles x 8 N-tiles = 16 tiles, 2 per wave; xw2 -> bufA[0..2687]
    for (int it = 0; it < 2; ++it) {
        int tile = wave + 8 * it;
        int mt = tile >> 3, nt = tile & 7;
        v8f acc = {0.f, 0.f, 0.f, 0.f, 0.f, 0.f, 0.f, 0.f};
        int arow = mt * 16 + l;
        int ar = (arow < NNODE) ? arow : 0;   // address clamp only
        int bcol = nt * 16 + l;
        const float* Abase = &bufB[ar * 256 + 2 * half];
        for (int k = 0; k < 256; k += 4) {
            v2f a = *(const v2f*)(Abase + k);          // ds_load_b64
            v2f bf;
            bf.x = W2[(k + 2 * half + 0) * 128 + bcol];
            bf.y = W2[(k + 2 * half + 1) * 128 + bcol];
            acc = __builtin_amdgcn_wmma_f32_16x16x4_f32(
                false, a, false, bf, (short)0, acc, false, false);
        }
        for (int g = 0; g < 8; ++g) {
            int row = mt * 16 + g + half * 8;
            if (row < NNODE) bufA[row * 128 + nt * 16 + l] = acc[g];
        }
    }
    __syncthreads();

    // ---- GAT2 logits (1 head, 128 ch) --------------------------------------
    if (t < NNODE) {
        float s = 0.f, d = 0.f;
        for (int c = 0; c < DMODEL; ++c) {
            float x = bufA[t * DMODEL + c];
            s += x * a_src2[c];
            d += x * a_dst2[c];
        }
        s2l[t] = s;
        d2l[t] = d;
    }
    __syncthreads();

    // ---- GAT2 softmax + aggregate -> h (bufA[2688..]) ----------------------
    if (t < NNODE) {
        int j = t;
        float dj = d2l[j];
        float ev[NNODE];
        float mx = -1e30f;
        for (int i = 0; i < NNODE; ++i) {
            float e = s2l[i] + dj;
            e = (e >= 0.f) ? e : 0.2f * e;
            bool m = (maskl[i * NNODE + j] != 0) || (i == j);
            ev[i] = m ? e : -1e30f;
            mx = fmaxf(mx, ev[i]);
        }
        float ssum = 0.f;
        for (int i = 0; i < NNODE; ++i) {
            float x = (ev[i] > -1e29f) ? expf(ev[i] - mx) : 0.f;
            ev[i] = x;
            ssum += x;
        }
        float rs = 1.f / ssum;
        for (int c = 0; c < DMODEL; ++c) {
            float acc = 0.f;
            for (int i = 0; i < NNODE; ++i)
                acc += ev[i] * bufA[i * DMODEL + c];
            bufA[NNODE * DMODEL + j * DMODEL + c] = acc * rs + b2[c];
        }
    }
    __syncthreads();

    // ---- readout MLP: hg1 = relu(h_flat @ Wr1 + br1) -----------------------
    {
        const float* hflat = &bufA[NNODE * DMODEL]; // 2688
        float acc = br1[t];
        for (int k = 0; k < NNODE * DMODEL; ++k)
            acc += hflat[k] * Wr1[k * 256 + t]; // coalesced across threads
        hg1[t] = (acc > 0.f) ? acc : 0.f;
    }
    __syncthreads();

    // ---- h_graph = hg1 @ Wr2 + br2 -----------------------------------------
    if (t < DMODEL) {
        float acc = br2[t];
        for (int k = 0; k < 256; ++k) acc += hg1[k] * Wr2[k * DMODEL + t];
        hgr[t] = acc;
    }
    __syncthreads();

    // ---- rank-1 collapse of final layer ------------------------------------
    // const[j] = h_graph@Wo[:128,j] + be@Wo[128:,j] + bo[j]
    // dir[j]   = We@Wo[128:,j]
    if (t < DMODEL) {
        float c = bo[t];
        for (int k = 0; k < DMODEL; ++k) c += hgr[k] * Wo[k * DMODEL + t];
        float dv = 0.f;
        for (int u = 0; u < 32; ++u) {
            float w = Wo[(DMODEL + u) * DMODEL + t];
            c += be[u] * w;
            dv += We[u] * w;
        }
        cvec[t] = c;
        dvec[t] = dv;
    }
}

// ---------------------------------------------------------------------------
// K3: out[b][j] = const[j] + me[b] * dir[j]   (pure streaming write)
// ---------------------------------------------------------------------------
__global__ void finalize_kernel(const float* __restrict__ me,
                                const float* __restrict__ cvec,
                                const float* __restrict__ dvec,
                                float* __restrict__ out,
                                int total) {
    int idx = blockIdx.x * blockDim.x + threadIdx.x;
    if (idx >= total) return;
    int b = idx >> 7;
    int j = idx & (DMODEL - 1);
    out[idx] = cvec[j] + me[b] * dvec[j];
}

// ---------------------------------------------------------------------------
extern "C" void kernel_launch(void* const* d_in, const int* in_sizes, int n_in,
                              void* d_out, int out_size, void* d_ws, size_t ws_size,
                              hipStream_t stream) {
    (void)n_in; (void)ws_size;
    const int* ids      = (const int*)d_in[0];
    const float* embed  = (const float*)d_in[1];
    const float* W1     = (const float*)d_in[2];
    const float* a_src1 = (const float*)d_in[3];
    const float* a_dst1 = (const float*)d_in[4];
    const float* b1     = (const float*)d_in[5];
    const float* W2     = (const float*)d_in[6];
    const float* a_src2 = (const float*)d_in[7];
    const float* a_dst2 = (const float*)d_in[8];
    const float* b2     = (const float*)d_in[9];
    const float* We     = (const float*)d_in[10];
    const float* be     = (const float*)d_in[11];
    const float* Wr1    = (const float*)d_in[12];
    const float* br1    = (const float*)d_in[13];
    const float* Wr2    = (const float*)d_in[14];
    const float* br2    = (const float*)d_in[15];
    const float* Wo     = (const float*)d_in[16];
    const float* bo     = (const float*)d_in[17];

    float* out = (float*)d_out;
    const int B = out_size / (DMODEL + 1); // h_mp (B*128) + me (B)
    const int S = in_sizes[0] / B;

    int* gmask  = (int*)d_ws;                          // 441 ints
    float* cvec = (float*)((char*)d_ws + 2048);        // 128 f
    float* dvec = cvec + DMODEL;                       // 128 f
    float* me   = out + (size_t)B * DMODEL;            // output tail

    zero_ws_kernel<<<1, 512, 0, stream>>>(gmask);
    hist_entropy_kernel<<<B, 256, 0, stream>>>(ids, me, gmask, S);
    graph_kernel<<<1, 256, 0, stream>>>(embed, W1, a_src1, a_dst1, b1,
                                        W2, a_src2, a_dst2, b2,
                                        We, be, Wr1, br1, Wr2, br2, Wo, bo,
                                        gmask, cvec, dvec);
    const int total = B * DMODEL;
    finalize_kernel<<<(total + 255) / 256, 256, 0, stream>>>(me, cvec, dvec, out, total);
}